// JGCF_Encoder_35003983462566
// MI455X (gfx1250) — compile-verified
//
#include <hip/hip_runtime.h>
#include <stdint.h>

typedef float v4f __attribute__((ext_vector_type(4)));  // clang vector: valid for NT stores

// ---------------------------------------------------------------------------
// CDNA5 async global->LDS staging helpers (gfx1250). ASYNCcnt is per-wave:
// each wave stages only its own 32 edge slots, so no workgroup barrier is
// needed between stage and consume; double-buffering uses s_wait_asynccnt 3
// (async loads complete in order, so <=3 outstanding means the current
// buffer's 3 loads have landed in LDS).
// ---------------------------------------------------------------------------
__device__ __forceinline__ void async_ld_b32(uint32_t lds_off, const void* gaddr) {
  asm volatile("global_load_async_to_lds_b32 %0, %1, off"
               :: "v"(lds_off), "v"(gaddr) : "memory");
}
__device__ __forceinline__ void wait_async_le3() {
  asm volatile("s_wait_asynccnt 0x3" ::: "memory");
}
__device__ __forceinline__ void wait_async_le0() {
  asm volatile("s_wait_asynccnt 0x0" ::: "memory");
}

// ---------------------------------------------------------------------------
// Coefficient kernel: alphas = 3*tanh(raw); Jacobi recurrence scalars.
// cbuf[lev*4 + {0,1,2}] = {sA (spmm), sB (x_cur), sC (x_prev)} for lev=1..3
// Runs almost entirely on the gfx1250 scalar-float ALU.
// ---------------------------------------------------------------------------
__global__ void k_coef(const float* __restrict__ alphas_raw, float* __restrict__ cbuf) {
  if (threadIdx.x != 0 || blockIdx.x != 0) return;
  const double a = 1.0, b = 1.0, l = -1.0, r = 1.0;
  float al[4];
#pragma unroll
  for (int i = 0; i < 4; ++i) al[i] = 3.0f * tanhf(alphas_raw[i]);
  // Level 1: x1 = coef1*x0 + coef2*spmm(x0)
  {
    double c1 = ((a - b) * 0.5 - (a + b + 2.0) * 0.5 * (l + r) / (r - l));
    double c2 = ((a + b + 2.0) / (r - l));
    cbuf[4 + 0] = (float)(c2 * (double)al[0]);
    cbuf[4 + 1] = (float)(c1 * (double)al[0]);
    cbuf[4 + 2] = 0.0f;
  }
  for (int Lev = 2; Lev <= 3; ++Lev) {
    double dl = (double)Lev;
    double coef_l    = 2.0 * dl * (dl + a + b) * (2.0 * dl - 2.0 + a + b);
    double coef_lm11 = (2.0 * dl + a + b - 1.0) * (2.0 * dl + a + b) * (2.0 * dl + a + b - 2.0);
    double coef_lm12 = (2.0 * dl + a + b - 1.0) * (a * a - b * b);
    double coef_lm2  = 2.0 * (dl - 1.0 + a) * (dl - 1.0 + b) * (2.0 * dl + a + b);
    double tmp1 = (double)al[Lev - 1] * (coef_lm11 / coef_l);
    double tmp2 = (double)al[Lev - 1] * (coef_lm12 / coef_l);
    double tmp3 = (double)al[Lev - 1] * (double)al[Lev - 2] * (coef_lm2 / coef_l);
    double tmp1_2 = tmp1 * (2.0 / (r - l));
    double tmp2_2 = tmp1 * ((r + l) / (r - l)) + tmp2;
    cbuf[Lev * 4 + 0] = (float)tmp1_2;
    cbuf[Lev * 4 + 1] = (float)(-tmp2_2);
    cbuf[Lev * 4 + 2] = (float)(-tmp3);
  }
}

// ---------------------------------------------------------------------------
// Virtual x0 (concat of user/item embeddings) v4f loader.
// i4 indexes float4 elements of the [N,64] array (16 per row).
// ---------------------------------------------------------------------------
__device__ __forceinline__ v4f ldx4(const float* buf, const float* xu, const float* xi,
                                    int is_x0, long i4, int U) {
  if (is_x0) {
    long n = i4 >> 4;
    int  k = (int)(i4 & 15);
    const float* base = (n < (long)U) ? (xu + n * 64) : (xi + (n - (long)U) * 64);
    return ((const v4f*)base)[k];
  }
  return ((const v4f*)buf)[i4];
}

// y = sB*cur + sC*prev   (vectorized v4f, grid-stride)
__global__ void k_axpby(float* __restrict__ y,
                        const float* cur, int cur_is_x0,
                        const float* prev, int prev_is_x0, int has_prev,
                        const float* __restrict__ xu, const float* __restrict__ xi,
                        const float* __restrict__ cbuf, int lev, int U, long n4) {
  const float sB = cbuf[lev * 4 + 1];
  const float sC = cbuf[lev * 4 + 2];
  long i      = (long)blockIdx.x * blockDim.x + threadIdx.x;
  long stride = (long)gridDim.x * blockDim.x;
  for (; i < n4; i += stride) {
    v4f c = ldx4(cur, xu, xi, cur_is_x0, i, U);
    v4f o;
    if (has_prev) {
      v4f p = ldx4(prev, xu, xi, prev_is_x0, i, U);
      o = sB * c + sC * p;
    } else {
      o = sB * c;
    }
    ((v4f*)y)[i] = o;
  }
}

// ---------------------------------------------------------------------------
// SpMM accumulate: y[row] += sA * val * x_cur[col], COO edges.
// Wave-per-32-edge chunks; edge triples staged to LDS via CDNA5 async
// global->LDS loads (double-buffered, per-wave ASYNCcnt sync). Two edges
// per inner step: half-wave per edge, one b128 gather (L2-resident) +
// 4 hardware f32 atomics per lane.
// ---------------------------------------------------------------------------
__global__ __launch_bounds__(256) void k_spmm(
    const float* __restrict__ vals, const int* __restrict__ rows, const int* __restrict__ cols,
    const float* __restrict__ cur, int cur_is_x0,
    const float* __restrict__ xu, const float* __restrict__ xi,
    float* __restrict__ y, const float* __restrict__ cbuf, int lev, int U, long nnz) {
  __shared__ float sv[2][8][32];
  __shared__ int   sr[2][8][32];
  __shared__ int   sc[2][8][32];

  const float sA   = cbuf[lev * 4 + 0];
  const int   lane = threadIdx.x & 31;
  const int   w    = threadIdx.x >> 5;
  const long  nchunks = (nnz + 31) >> 5;
  const long  nw = (long)gridDim.x * (blockDim.x >> 5);
  long gw = (long)blockIdx.x * (blockDim.x >> 5) + w;

  const int half = lane >> 4;   // which edge of the pair this lane serves
  const int h    = lane & 15;   // float4 slot within the 64-float row

  int buf = 0;
  if (gw < nchunks) {
    long e = gw * 32 + lane;
    if (e < nnz) {
      async_ld_b32((uint32_t)(uintptr_t)&sv[0][w][lane], vals + e);
      async_ld_b32((uint32_t)(uintptr_t)&sr[0][w][lane], rows + e);
      async_ld_b32((uint32_t)(uintptr_t)&sc[0][w][lane], cols + e);
    }
  }

  for (long c = gw; c < nchunks; c += nw) {
    long cn = c + nw;
    if (cn < nchunks) {
      int  nb = buf ^ 1;
      long e  = cn * 32 + lane;
      if (e < nnz) {
        async_ld_b32((uint32_t)(uintptr_t)&sv[nb][w][lane], vals + e);
        async_ld_b32((uint32_t)(uintptr_t)&sr[nb][w][lane], rows + e);
        async_ld_b32((uint32_t)(uintptr_t)&sc[nb][w][lane], cols + e);
      }
      wait_async_le3();   // current buffer's 3 loads complete (in-order)
    } else {
      wait_async_le0();   // drain everything
    }

    long rem  = nnz - c * 32;
    int  jmax = rem > 32 ? 32 : (int)rem;

#pragma unroll 4
    for (int j = 0; j < 32; j += 2) {
      int j2 = j + half;
      if (j2 < jmax) {
        float v  = sv[buf][w][j2];
        int   rr = sr[buf][w][j2];
        int   cc = sc[buf][w][j2];
        const float* xb;
        if (cur_is_x0)
          xb = (cc < U) ? (xu + (long)cc * 64) : (xi + ((long)cc - U) * 64);
        else
          xb = cur + (long)cc * 64;
        v4f   p = ((const v4f*)xb)[h];
        float s = sA * v;
        float* yp = y + (long)rr * 64 + h * 4;
        unsafeAtomicAdd(yp + 0, s * p.x);
        unsafeAtomicAdd(yp + 1, s * p.y);
        unsafeAtomicAdd(yp + 2, s * p.z);
        unsafeAtomicAdd(yp + 3, s * p.w);
      }
    }
    buf ^= 1;
  }
}

// low = (x0+x1+x2+x3)/4 ; mid = 0.1*x0 - low ; out[n] = [low | mid] (128 floats)
// Output is write-once: non-temporal stores keep it from displacing the
// L2-resident x-layers / edge arrays.
__global__ void k_final(float* __restrict__ out,
                        const float* __restrict__ x1, const float* __restrict__ x2,
                        const float* __restrict__ x3,
                        const float* __restrict__ xu, const float* __restrict__ xi,
                        int U, long n4) {
  long i      = (long)blockIdx.x * blockDim.x + threadIdx.x;
  long stride = (long)gridDim.x * blockDim.x;
  for (; i < n4; i += stride) {
    long n = i >> 4;
    int  k = (int)(i & 15);
    v4f x0 = ldx4(nullptr, xu, xi, 1, i, U);
    v4f a  = ((const v4f*)x1)[i];
    v4f b  = ((const v4f*)x2)[i];
    v4f c  = ((const v4f*)x3)[i];
    v4f low = 0.25f * (x0 + a + b + c);
    v4f mid = 0.1f * x0 - low;
    __builtin_nontemporal_store(low, (v4f*)out + n * 32 + k);
    __builtin_nontemporal_store(mid, (v4f*)out + n * 32 + 16 + k);
  }
}

extern "C" void kernel_launch(void* const* d_in, const int* in_sizes, int n_in,
                              void* d_out, int out_size, void* d_ws, size_t ws_size,
                              hipStream_t stream) {
  const float* user = (const float*)d_in[0];
  const float* item = (const float*)d_in[1];
  const float* vals = (const float*)d_in[2];
  const float* araw = (const float*)d_in[3];
  const int*   rows = (const int*)d_in[4];
  const int*   cols = (const int*)d_in[5];

  const int  U    = in_sizes[0] / 64;
  const int  Itm  = in_sizes[1] / 64;
  const long N    = (long)U + (long)Itm;
  const long nnz  = (long)in_sizes[2];
  float* out = (float*)d_out;

  char*  ws   = (char*)d_ws;
  float* cbuf = (float*)ws;                       // 64 floats of coefficients
  float* x1   = (float*)(ws + 256);
  float* x2   = x1 + N * 64;
  float* x3   = x2 + N * 64;
  (void)ws_size; (void)n_in; (void)out_size;

  const long n4 = N * 16;
  const int  tb = 256;
  int ew_blocks = (int)((n4 + tb - 1) / tb);
  if (ew_blocks > 8192) ew_blocks = 8192;
  const int sp_blocks = 3072;

  k_coef<<<dim3(1), dim3(32), 0, stream>>>(araw, cbuf);

  // Layer 1: x1 = sB*x0, then x1 += sA * A x0
  k_axpby<<<ew_blocks, tb, 0, stream>>>(x1, nullptr, 1, nullptr, 1, 0, user, item, cbuf, 1, U, n4);
  k_spmm <<<sp_blocks, 256, 0, stream>>>(vals, rows, cols, nullptr, 1, user, item, x1, cbuf, 1, U, nnz);

  // Layer 2: x2 = sB*x1 + sC*x0, then x2 += sA * A x1
  k_axpby<<<ew_blocks, tb, 0, stream>>>(x2, x1, 0, nullptr, 1, 1, user, item, cbuf, 2, U, n4);
  k_spmm <<<sp_blocks, 256, 0, stream>>>(vals, rows, cols, x1, 0, user, item, x2, cbuf, 2, U, nnz);

  // Layer 3: x3 = sB*x2 + sC*x1, then x3 += sA * A x2
  k_axpby<<<ew_blocks, tb, 0, stream>>>(x3, x2, 0, x1, 0, 1, user, item, cbuf, 3, U, n4);
  k_spmm <<<sp_blocks, 256, 0, stream>>>(vals, rows, cols, x2, 0, user, item, x3, cbuf, 3, U, nnz);

  k_final<<<ew_blocks, tb, 0, stream>>>(out, x1, x2, x3, user, item, U, n4);
}